// VLineFPNFeatureExtractor_21509196218381
// MI455X (gfx1250) — compile-verified
//
#include <hip/hip_runtime.h>
#include <math.h>
#include <stdint.h>

typedef __attribute__((ext_vector_type(2))) float v2f;
typedef __attribute__((ext_vector_type(8))) float v8f;

#define RESP   7
#define NSAMP  14      // RES*SR samples per axis
#define KENT   28      // 2*NSAMP (row,weight) entries per axis
#define TILE_C 32      // padded column count (cols 28..31 duplicate col 27)
#define UB_STRIDE 34   // Ubuf row stride in floats (8B-aligned, conflict-free)
#define NWAVE  8
#define NCH    256

struct SharedMem {
    float Ry[16][KENT];               // y-axis weights, rows 7..15 zero
    float Rx[16][KENT];               // x-axis weights, rows 7..15 zero
    int   rows[KENT];
    int   colsPad[TILE_C];            // 28 real cols + 4 duplicated pads
    float T[NWAVE][2][TILE_C * KENT]; // per-wave double-buffered tile, COLUMN-major [col][row]
    float U[NWAVE][16 * UB_STRIDE];   // per-wave stage-1 re-layout buffer
};

// Issue 28 async global->LDS copies for one channel tile. Lane owns one padded
// column; iteration t walks rows. GVS mode: wave-uniform SGPR base + per-lane
// 32-bit byte offset. LDS dest is column-major: base + lane*112 + t*4.
__device__ __forceinline__ void issue_async_tile(const float* cp, unsigned ldsBase,
                                                 const int (&goffB)[KENT]) {
    unsigned long long a = (unsigned long long)(uintptr_t)cp;   // wave-uniform
    unsigned lo = __builtin_amdgcn_readfirstlane((unsigned)a);
    unsigned hi = __builtin_amdgcn_readfirstlane((unsigned)(a >> 32));
    unsigned long long sb = ((unsigned long long)hi << 32) | lo;
#pragma unroll
    for (int t = 0; t < KENT; ++t) {
        unsigned lds = ldsBase + t * 4;
        asm volatile("global_load_async_to_lds_b32 %0, %1, %2"
                     :: "v"(lds), "v"(goffB[t]), "s"(sb)
                     : "memory");
    }
}

__global__ __launch_bounds__(256) void roi_align_fpn_wmma(
    const float* __restrict__ f0, const float* __restrict__ f1,
    const float* __restrict__ f2, const float* __restrict__ f3,
    const float* __restrict__ boxes, const int* __restrict__ batch_idx,
    float* __restrict__ out)
{
    __shared__ SharedMem sm;
    const int r    = blockIdx.x;
    const int tid  = threadIdx.x;
    const int lane = tid & 31;
    const int wid  = tid >> 5;
    const int hi   = lane >> 4;   // lane half (selects K pair in fragments)
    const int mn   = lane & 15;   // M (A-frag) or N (B/C/D-frag) index

    // ---- per-ROI level selection (uniform across block) ----
    const float x1b = boxes[r * 4 + 0], y1b = boxes[r * 4 + 1];
    const float x2b = boxes[r * 4 + 2], y2b = boxes[r * 4 + 3];
    const float area = fmaxf(x2b - x1b, 0.0f) * fmaxf(y2b - y1b, 0.0f);
    const float sdim = sqrtf(area);
    float lvlf = floorf(4.0f + log2f(sdim / 224.0f + 1e-6f));
    lvlf = fminf(fmaxf(lvlf, 2.0f), 5.0f);
    const int   k     = (int)lvlf - 2;          // 0..3
    const int   H     = 200 >> k;
    const int   W     = H;
    const float scale = 0.25f / (float)(1 << k);
    const int   b     = batch_idx[r];
    const float* fsel = (k == 0) ? f0 : (k == 1) ? f1 : (k == 2) ? f2 : f3;
    const size_t HW   = (size_t)H * (size_t)W;
    const float* featBase = fsel + (size_t)b * NCH * HW;

    // ---- phase 0a: zero weight matrices ----
    {
        float* ryf = &sm.Ry[0][0];
        float* rxf = &sm.Rx[0][0];
        for (int idx = tid; idx < 16 * KENT; idx += 256) { ryf[idx] = 0.0f; rxf[idx] = 0.0f; }
    }
    __syncthreads();

    // ---- phase 0b: per-axis sample indices + weights ----
    if (tid < NSAMP) {                       // y axis, sample m = tid
        const int m = tid;
        const float y1s  = y1b * scale;
        const float binh = fmaxf(y2b * scale - y1s, 1.0f) * (1.0f / RESP);
        const float p    = (float)(m >> 1) + 0.25f + 0.5f * (float)(m & 1);
        const float yv   = y1s + p * binh;
        const float vm   = (yv >= -1.0f && yv <= (float)H) ? 0.5f : 0.0f;
        const float yc   = fminf(fmaxf(yv, 0.0f), (float)(H - 1));
        const int   y0   = (int)floorf(yc);
        const int   y1i  = min(y0 + 1, H - 1);
        const float ly   = yc - (float)y0;
        sm.rows[2 * m]     = y0;
        sm.rows[2 * m + 1] = y1i;
        sm.Ry[m >> 1][2 * m]     = vm * (1.0f - ly);
        sm.Ry[m >> 1][2 * m + 1] = vm * ly;
    } else if (tid >= 16 && tid < 16 + NSAMP) {  // x axis, sample m = tid-16
        const int m = tid - 16;
        const float x1s  = x1b * scale;
        const float binw = fmaxf(x2b * scale - x1s, 1.0f) * (1.0f / RESP);
        const float p    = (float)(m >> 1) + 0.25f + 0.5f * (float)(m & 1);
        const float xv   = x1s + p * binw;
        const float vm   = (xv >= -1.0f && xv <= (float)W) ? 0.5f : 0.0f;
        const float xc   = fminf(fmaxf(xv, 0.0f), (float)(W - 1));
        const int   x0   = (int)floorf(xc);
        const int   x1i  = min(x0 + 1, W - 1);
        const float lx   = xc - (float)x0;
        sm.colsPad[2 * m]     = x0;
        sm.colsPad[2 * m + 1] = x1i;
        sm.Rx[m >> 1][2 * m]     = vm * (1.0f - lx);
        sm.Rx[m >> 1][2 * m + 1] = vm * lx;
        if (m == NSAMP - 1) {                 // fill pad columns 28..31
            sm.colsPad[28] = x1i; sm.colsPad[29] = x1i;
            sm.colsPad[30] = x1i; sm.colsPad[31] = x1i;
        }
    }
    __syncthreads();

    // ---- resident A(Ry) and B(Rx^T) fragments: 7 K-chunks of 4, f32 16x16x4 layout ----
    v2f aY[7], bX[7];
#pragma unroll
    for (int kk = 0; kk < 7; ++kk) {
        aY[kk].x = sm.Ry[mn][4 * kk + 2 * hi];
        aY[kk].y = sm.Ry[mn][4 * kk + 2 * hi + 1];
        bX[kk].x = sm.Rx[mn][4 * kk + 2 * hi];
        bX[kk].y = sm.Rx[mn][4 * kk + 2 * hi + 1];
    }

    // ---- per-lane gather byte-offsets (channel-invariant): lane = column, t = row ----
    int goffB[KENT];
    {
        const int colv = sm.colsPad[lane];
#pragma unroll
        for (int t = 0; t < KENT; ++t)
            goffB[t] = (sm.rows[t] * W + colv) * 4;
    }

    float* Ubuf = &sm.U[wid][0];
    const unsigned ldsT0 = (unsigned)(uintptr_t)&sm.T[wid][0][0] + lane * (KENT * 4);
    const unsigned ldsT1 = (unsigned)(uintptr_t)&sm.T[wid][1][0] + lane * (KENT * 4);
    const v8f vzero = {0.f, 0.f, 0.f, 0.f, 0.f, 0.f, 0.f, 0.f};
    const int cbase = wid << 5;

    // prologue: prefetch first channel of this wave into buffer 0
    issue_async_tile(featBase + (size_t)cbase * HW, ldsT0, goffB);

    // ---- channel loop: each wave owns 32 channels, double-buffered pipeline ----
#pragma unroll 1
    for (int cc = 0; cc < 32; ++cc) {
        const int cur = cc & 1;
        const int c   = cbase + cc;

        if (cc < 31) {
            // prefetch next channel into the alternate buffer
            issue_async_tile(featBase + (size_t)(c + 1) * HW,
                             cur ? ldsT0 : ldsT1, goffB);
            // async loads complete in order: <=28 outstanding => current tile landed
            asm volatile("s_wait_asynccnt 0x1c" ::: "memory");
        } else {
            asm volatile("s_wait_asynccnt 0x0" ::: "memory");
        }

        const float* Ttile = &sm.T[wid][cur][0];   // column-major [col][row]

        // stage 1: U(16 x 32) = Ry(16x28) * T(28x32), 7 K-chunks x 2 N-tiles.
        // B-fragment pair {T[rrow][col], T[rrow+1][col]} is contiguous in LDS
        // (column-major) -> single aligned b64 load per fragment, no repacking.
        v8f U0 = vzero, U1 = vzero;
#pragma unroll
        for (int kk = 0; kk < 7; ++kk) {
            const int rrow = 4 * kk + 2 * hi;
            const v2f b0 = *(const v2f*)(Ttile + mn * KENT + rrow);
            const v2f b1 = *(const v2f*)(Ttile + (16 + mn) * KENT + rrow);
            U0 = __builtin_amdgcn_wmma_f32_16x16x4_f32(false, aY[kk], false, b0,
                                                       (short)0, U0, false, false);
            U1 = __builtin_amdgcn_wmma_f32_16x16x4_f32(false, aY[kk], false, b1,
                                                       (short)0, U1, false, false);
        }

        // re-layout U: D-fragments -> LDS (16 x UB_STRIDE) -> A-fragments
#pragma unroll
        for (int d = 0; d < 8; ++d) {
            Ubuf[(d + 8 * hi) * UB_STRIDE + mn]      = U0[d];
            Ubuf[(d + 8 * hi) * UB_STRIDE + 16 + mn] = U1[d];
        }

        // stage 2: out(16x16) = U(16x28) * Rx^T(28x16), 7 K-chunks
        v8f O = vzero;
#pragma unroll
        for (int kk = 0; kk < 7; ++kk) {
            const v2f a2 = *(const v2f*)(Ubuf + mn * UB_STRIDE + 4 * kk + 2 * hi);
            O = __builtin_amdgcn_wmma_f32_16x16x4_f32(false, a2, false, bX[kk],
                                                      (short)0, O, false, false);
        }

        // D layout: lanes 0-15 hold M=d (vgpr d), N=lane; only ph<7, pw<7 are real
        if (lane < RESP) {
            float* op = out + ((size_t)r * NCH + c) * (RESP * RESP);
#pragma unroll
            for (int d = 0; d < RESP; ++d) op[d * RESP + lane] = O[d];
        }
    }
}

extern "C" void kernel_launch(void* const* d_in, const int* in_sizes, int n_in,
                              void* d_out, int out_size, void* d_ws, size_t ws_size,
                              hipStream_t stream) {
    const float* f0    = (const float*)d_in[0];
    const float* f1    = (const float*)d_in[1];
    const float* f2    = (const float*)d_in[2];
    const float* f3    = (const float*)d_in[3];
    const float* boxes = (const float*)d_in[4];
    const int*   bidx  = (const int*)d_in[5];
    float* out = (float*)d_out;
    const int R = in_sizes[4] / 4;  // 512 ROIs
    roi_align_fpn_wmma<<<R, 256, 0, stream>>>(f0, f1, f2, f3, boxes, bidx, out);
}